// ChamferDistance_9689446220275
// MI455X (gfx1250) — compile-verified
//
#include <hip/hip_runtime.h>
#include <float.h>
#include <math.h>

typedef float v2f __attribute__((ext_vector_type(2)));
typedef float v8f __attribute__((ext_vector_type(8)));

// One wave owns TWO 16-row tiles of P (32 rows, for one batch) and scans all
// 16-col tiles of Q. The K=3 slot of the 16x16x4 f32 WMMA folds the column
// norm into the product:
//   A row = (px, py, pz, 1)
//   B col = (-2qx, -2qy, -2qz, q^2)
//   =>  C[m,n] = q_n^2 - 2 p_m . q_n
// Each B tile is built once and consumed by two WMMAs (one per A tile), so
// the per-WMMA cost is ~0.5 load + 3 VALU + 4 min3.
// Row norm p^2 (constant per row) is added after the column min-reduction;
// max(.,0) clamping commutes with min and is applied at writeout.
__global__ __launch_bounds__(256) void chamfer_min_d2_kernel(
    const float* __restrict__ P,    // (B, NP, 3)
    const float* __restrict__ Q,    // (B, NQ, 3)
    float* __restrict__ outMinD2,   // (B, NP)
    int NP, int NQ)
{
    const int lane    = threadIdx.x & 31;
    const int waveIn  = threadIdx.x >> 5;               // 8 waves / block
    const int wave    = blockIdx.x * 8 + waveIn;        // 32-row strip index
    const int b       = blockIdx.y;                     // batch

    const float* Pb = P + (size_t)b * NP * 3;
    const float* Qb = Q + (size_t)b * NQ * 3;

    const bool hi  = (lane >= 16);
    const int  r16 = lane & 15;
    const int  base = wave << 5;                        // first of 32 rows

    // ---- Two A tiles (rows base..base+15 and base+16..base+31).
    // A layout (ISA 7.12.2, 32-bit A 16x4): v0: lanes0-15 K=0, lanes16-31 K=2
    //                                       v1: lanes0-15 K=1, lanes16-31 K=3
    const int prow0 = base + r16;
    const int prow1 = base + 16 + r16;
    const float p0x = Pb[prow0 * 3 + 0];
    const float p0y = Pb[prow0 * 3 + 1];
    const float p0z = Pb[prow0 * 3 + 2];
    const float p1x = Pb[prow1 * 3 + 0];
    const float p1y = Pb[prow1 * 3 + 1];
    const float p1z = Pb[prow1 * 3 + 2];
    v2f a0, a1;
    a0.x = hi ? p0z : p0x;
    a0.y = hi ? 1.0f : p0y;      // K=3 slot multiplies B's q^2 term
    a1.x = hi ? p1z : p1x;
    a1.y = hi ? 1.0f : p1y;

    // ||p||^2 per lane for row (lane&15) of each tile; broadcast into the C
    // layout: C vgpr v, lane-half h -> row m = v + 8*h.
    const float p2lane0 = p0x * p0x + p0y * p0y + p0z * p0z;
    const float p2lane1 = p1x * p1x + p1y * p1y + p1z * p1z;
    float x2c0[8], x2c1[8];
#pragma unroll
    for (int v = 0; v < 8; ++v) {
        const int src = hi ? (v + 8) : v;
        x2c0[v] = __shfl(p2lane0, src, 32);
        x2c1[v] = __shfl(p2lane1, src, 32);
    }

    float minv0[8], minv1[8];
#pragma unroll
    for (int v = 0; v < 8; ++v) { minv0[v] = FLT_MAX; minv1[v] = FLT_MAX; }

    const int tilesQ = NQ >> 4;
#pragma unroll 2
    for (int t = 0; t < tilesQ; ++t) {
        // ---- B tile (16 q-points), column n = lane&15.
        // B layout mirrors A: v0: lanes0-15 K=0, lanes16-31 K=2
        //                     v1: lanes0-15 K=1, lanes16-31 K=3
        const int qrow = (t << 4) + r16;
        const float qx = Qb[qrow * 3 + 0];
        const float qy = Qb[qrow * 3 + 1];
        const float qz = Qb[qrow * 3 + 2];
        const float q2 = qx * qx + qy * qy + qz * qz;
        v2f bm;
        bm.x = -2.0f * (hi ? qz : qx);
        bm.y = hi ? q2 : (-2.0f * qy);

        v8f c0 = {}, c1 = {};
        c0 = __builtin_amdgcn_wmma_f32_16x16x4_f32(
            false, a0, false, bm, (short)0, c0, false, false);
        c1 = __builtin_amdgcn_wmma_f32_16x16x4_f32(
            false, a1, false, bm, (short)0, c1, false, false);

#pragma unroll
        for (int v = 0; v < 8; ++v) {
            minv0[v] = fminf(minv0[v], c0[v]);   // running min of (q^2 - 2 x.q)
            minv1[v] = fminf(minv1[v], c1[v]);
        }
    }

    // ---- Reduce across the 16 columns (lanes within each half); xor masks
    // 1,2,4,8 never cross the half boundary in wave32.
#pragma unroll
    for (int v = 0; v < 8; ++v) {
        float m0 = minv0[v];
        m0 = fminf(m0, __shfl_xor(m0, 1, 32));
        m0 = fminf(m0, __shfl_xor(m0, 2, 32));
        m0 = fminf(m0, __shfl_xor(m0, 4, 32));
        m0 = fminf(m0, __shfl_xor(m0, 8, 32));
        minv0[v] = m0;
        float m1 = minv1[v];
        m1 = fminf(m1, __shfl_xor(m1, 1, 32));
        m1 = fminf(m1, __shfl_xor(m1, 2, 32));
        m1 = fminf(m1, __shfl_xor(m1, 4, 32));
        m1 = fminf(m1, __shfl_xor(m1, 8, 32));
        minv1[v] = m1;
    }

    // Lane 0 holds rows 0..7 of each tile, lane 16 holds rows 8..15.
    // Add the row norm now, clamp at 0 (commutes with min).
    if (lane == 0 || lane == 16) {
        float* o0 = outMinD2 + (size_t)b * NP + base + (hi ? 8 : 0);
        float* o1 = o0 + 16;
#pragma unroll
        for (int v = 0; v < 8; ++v) {
            o0[v] = fmaxf(minv0[v] + x2c0[v], 0.0f);
            o1[v] = fmaxf(minv1[v] + x2c1[v], 0.0f);
        }
    }
}

// Single-block final reduction:
// loss = mean(sqrt(minP2)) + bidir * mean(sqrt(minQ2)); the per-batch mean and
// batch mean fold into one global mean since all batches are equal-sized.
__global__ __launch_bounds__(1024) void chamfer_final_kernel(
    const float* __restrict__ minP2,  // B*NP
    const float* __restrict__ minQ2,  // B*NQ
    const int*  __restrict__ bidir,
    float* __restrict__ out,
    int lenP, int lenQ)
{
    __shared__ float sbuf[1024];
    float accP = 0.0f, accQ = 0.0f;
    for (int i = threadIdx.x; i < lenP; i += blockDim.x) accP += sqrtf(minP2[i]);
    for (int i = threadIdx.x; i < lenQ; i += blockDim.x) accQ += sqrtf(minQ2[i]);

    float v = accP / (float)lenP;
    if (bidir[0] != 0) v += accQ / (float)lenQ;

    sbuf[threadIdx.x] = v;
    __syncthreads();
    for (int s = 512; s > 0; s >>= 1) {
        if ((int)threadIdx.x < s) sbuf[threadIdx.x] += sbuf[threadIdx.x + s];
        __syncthreads();
    }
    if (threadIdx.x == 0) out[0] = sbuf[0];
}

extern "C" void kernel_launch(void* const* d_in, const int* in_sizes, int n_in,
                              void* d_out, int out_size, void* d_ws, size_t ws_size,
                              hipStream_t stream) {
    (void)n_in; (void)out_size; (void)ws_size;

    const float* x     = (const float*)d_in[0];   // (B, N, 3)
    const float* y     = (const float*)d_in[1];   // (B, M, 3)
    const int*   bidir = (const int*)d_in[2];

    const int B = 4, D = 3;
    const int N = in_sizes[0] / (B * D);
    const int M = in_sizes[1] / (B * D);

    float* minP2 = (float*)d_ws;                  // B*N floats
    float* minQ2 = minP2 + (size_t)B * N;         // B*M floats

    // Pass 1: for each x, min over y. One wave per 32 rows.
    dim3 grid1(N / 32 / 8, B);
    chamfer_min_d2_kernel<<<grid1, 256, 0, stream>>>(x, y, minP2, N, M);

    // Pass 2 (bidirectional term): for each y, min over x. Always computed;
    // inclusion gated on-device by the bidirectional flag.
    dim3 grid2(M / 32 / 8, B);
    chamfer_min_d2_kernel<<<grid2, 256, 0, stream>>>(y, x, minQ2, M, N);

    chamfer_final_kernel<<<1, 1024, 0, stream>>>(minP2, minQ2, bidir,
                                                 (float*)d_out, B * N, B * M);
}